// CDA_58377195487596
// MI455X (gfx1250) — compile-verified
//
#include <hip/hip_runtime.h>

#define B_  16
#define C_  256
#define R_  16
#define H_  128
#define W_  128
#define HW_ (H_*W_)   // 16384

typedef __attribute__((ext_vector_type(2))) float v2f;
typedef __attribute__((ext_vector_type(8))) float v8f;

__device__ __forceinline__ float sigmoidf_(float v) { return 1.0f / (1.0f + expf(-v)); }

// ---------------------------------------------------------------------------
// Kernel 1: global avg + max pool over H*W per (b,c).  One block per (b,c).
// ---------------------------------------------------------------------------
__global__ void __launch_bounds__(256)
pool_kernel(const float* __restrict__ x,
            float* __restrict__ pavg, float* __restrict__ pmax) {
  __shared__ float ssum[256];
  __shared__ float smax[256];
  const int bc  = blockIdx.x;
  const int tid = threadIdx.x;
  const float4* xp = reinterpret_cast<const float4*>(x + ((size_t)bc << 14));
  float s = 0.0f, m = -__builtin_inff();
  #pragma unroll 4
  for (int i = tid; i < HW_ / 4; i += 256) {
    float4 v = xp[i];
    s += v.x + v.y + v.z + v.w;
    m = fmaxf(m, fmaxf(fmaxf(v.x, v.y), fmaxf(v.z, v.w)));
  }
  ssum[tid] = s; smax[tid] = m;
  __syncthreads();
  for (int off = 128; off > 0; off >>= 1) {
    if (tid < off) {
      ssum[tid] += ssum[tid + off];
      smax[tid]  = fmaxf(smax[tid], smax[tid + off]);
    }
    __syncthreads();
  }
  if (tid == 0) {
    pavg[bc] = ssum[0] * (1.0f / HW_);
    pmax[bc] = smax[0];
  }
}

// ---------------------------------------------------------------------------
// Kernel 2: channel MLP with fp32 WMMA (single wave32, EXEC all ones).
//   GEMM1: [16,256] x [256,16] (for avg and max, shared B fragments)
//   relu+bias, sum; GEMM2: [16,16] x [16,256]; sigmoid -> wch[16,256].
// WMMA layouts per CDNA5 ISA 7.12.2 (32-bit A 16x4, 32-bit C/D 16x16).
// ---------------------------------------------------------------------------
__global__ void __launch_bounds__(32)
mlp_wmma_kernel(const float* __restrict__ pavg, const float* __restrict__ pmax,
                const float* __restrict__ w1, const float* __restrict__ b1,
                const float* __restrict__ w2, const float* __restrict__ b2,
                float* __restrict__ wch) {
  const int lane = threadIdx.x;   // 0..31
  const int col  = lane & 15;     // A: M-row / B: N-col / C,D: N-col
  const int g    = lane >> 4;     // lane group selects K pair / M half

  // ---- GEMM1 over K=256, two accumulator chains sharing B ----
  v8f acc_a = {}; v8f acc_m = {};
  for (int k0 = 0; k0 < C_; k0 += 4) {
    const int ka = k0 + 2 * g;
    v2f Aa, Am, Bb;
    Aa[0] = pavg[col * C_ + ka];  Aa[1] = pavg[col * C_ + ka + 1];
    Am[0] = pmax[col * C_ + ka];  Am[1] = pmax[col * C_ + ka + 1];
    // B = w1^T : B[k][n] = w1[n*C_ + k]
    Bb[0] = w1[col * C_ + ka];    Bb[1] = w1[col * C_ + ka + 1];
    acc_a = __builtin_amdgcn_wmma_f32_16x16x4_f32(false, Aa, false, Bb,
                                                  (short)0, acc_a, false, false);
    acc_m = __builtin_amdgcn_wmma_f32_16x16x4_f32(false, Am, false, Bb,
                                                  (short)0, acc_m, false, false);
  }

  // bias + relu, sum the two hidden activations; restage D-layout -> A-layout
  __shared__ float hmat[16][16];
  const float bias1 = b1[col];
  #pragma unroll
  for (int i = 0; i < 8; ++i) {
    const int row = i + 8 * g;            // batch index
    const float ha = fmaxf(acc_a[i] + bias1, 0.0f);
    const float hm = fmaxf(acc_m[i] + bias1, 0.0f);
    hmat[row][col] = ha + hm;
  }
  __syncthreads();

  // ---- GEMM2: h[16,16] x w2^T[16,256] ----
  v2f Af[4];
  #pragma unroll
  for (int t = 0; t < 4; ++t) {
    const int ka = 4 * t + 2 * g;
    Af[t][0] = hmat[col][ka];
    Af[t][1] = hmat[col][ka + 1];
  }
  for (int tile = 0; tile < 16; ++tile) {
    const int n = tile * 16 + col;        // output channel
    v8f acc = {};
    #pragma unroll
    for (int t = 0; t < 4; ++t) {
      const int ka = 4 * t + 2 * g;
      v2f Bb;                             // B = w2^T : B[k][n] = w2[n*R_ + k]
      Bb[0] = w2[n * R_ + ka];
      Bb[1] = w2[n * R_ + ka + 1];
      acc = __builtin_amdgcn_wmma_f32_16x16x4_f32(false, Af[t], false, Bb,
                                                  (short)0, acc, false, false);
    }
    const float bias2 = 2.0f * b2[n];     // mlp(avg)+mlp(mx) each add b2
    #pragma unroll
    for (int i = 0; i < 8; ++i) {
      const int row = i + 8 * g;
      wch[row * C_ + n] = sigmoidf_(acc[i] + bias2);
    }
  }
}

// ---------------------------------------------------------------------------
// Kernel 3: y = x*wch; channel mean/max -> cat[b][0]=maxmap, cat[b][1]=avgmap
// ---------------------------------------------------------------------------
__global__ void __launch_bounds__(256)
ymap_kernel(const float* __restrict__ x, const float* __restrict__ wch,
            float* __restrict__ cat) {
  __shared__ float wc[C_];
  const int tid = threadIdx.x;
  const int b   = blockIdx.x >> 6;                    // 64 blocks per batch
  const int hw  = ((blockIdx.x & 63) << 8) + tid;     // 256 pixels per block
  wc[tid] = wch[b * C_ + tid];
  __syncthreads();
  const float* xb = x + (((size_t)b * C_) << 14);
  float s = 0.0f, m = -__builtin_inff();
  for (int c = 0; c < C_; ++c) {
    if (c + 8 < C_)
      __builtin_prefetch(&xb[((size_t)(c + 8) << 14) + hw], 0, 1);
    const float v = xb[((size_t)c << 14) + hw] * wc[c];
    s += v;
    m = fmaxf(m, v);
  }
  float* cb = cat + ((size_t)b * 2) * HW_;
  cb[hw]       = m;                 // channel 0 = maxmap
  cb[HW_ + hw] = s * (1.0f / C_);   // channel 1 = avgmap
}

// ---------------------------------------------------------------------------
// Kernel 4: offset conv (2->18, 3x3 SAME) + deformable conv (2->1) + BN + sig
// ---------------------------------------------------------------------------
__device__ __forceinline__ float samp2d(const float* __restrict__ mp, int y, int x) {
  const bool valid = (y >= 0) & (y < H_) & (x >= 0) & (x < W_);
  const int cy = min(max(y, 0), H_ - 1);
  const int cx = min(max(x, 0), W_ - 1);
  return valid ? mp[cy * W_ + cx] : 0.0f;
}

__global__ void __launch_bounds__(256)
dsa_kernel(const float* __restrict__ cat,
           const float* __restrict__ off_w, const float* __restrict__ off_b,
           const float* __restrict__ dc_w,  const float* __restrict__ dc_b,
           const float* __restrict__ bn_gamma, const float* __restrict__ bn_beta,
           const float* __restrict__ bn_mean,  const float* __restrict__ bn_var,
           float* __restrict__ wsmap) {
  __shared__ float s_ow[18 * 2 * 9];
  __shared__ float s_ob[18];
  __shared__ float s_dw[18];
  const int tid = threadIdx.x;
  for (int i = tid; i < 18 * 2 * 9; i += 256) s_ow[i] = off_w[i];
  if (tid < 18) { s_ob[tid] = off_b[tid]; s_dw[tid] = dc_w[tid]; }
  __syncthreads();

  const int p  = blockIdx.x * 256 + tid;
  const int b  = p >> 14;
  const int hw = p & (HW_ - 1);
  const int h  = hw >> 7;
  const int w  = hw & (W_ - 1);
  const float* cm = cat + ((size_t)b * 2) * HW_;  // channel 0 (maxmap)
  const float* ca = cm + HW_;                     // channel 1 (avgmap)

  // offsets = conv3x3(cat) + bias  (18 output channels, zero-padded SAME)
  float offv[18];
  #pragma unroll
  for (int ch = 0; ch < 18; ++ch) offv[ch] = s_ob[ch];
  #pragma unroll
  for (int i = 0; i < 2; ++i) {
    const float* cin = i ? ca : cm;
    #pragma unroll
    for (int ky = 0; ky < 3; ++ky) {
      const int yy = h + ky - 1;
      if (yy < 0 || yy >= H_) continue;
      #pragma unroll
      for (int kx = 0; kx < 3; ++kx) {
        const int xx = w + kx - 1;
        if (xx < 0 || xx >= W_) continue;
        const float v = cin[yy * W_ + xx];
        #pragma unroll
        for (int ch = 0; ch < 18; ++ch)
          offv[ch] += v * s_ow[(ch * 2 + i) * 9 + ky * 3 + kx];
      }
    }
  }

  // deformable 3x3 conv with bilinear sampling (zero outside, clamp gather)
  float acc = dc_b[0];
  #pragma unroll
  for (int k = 0; k < 9; ++k) {
    const float py = (float)h + (float)(k / 3 - 1) + offv[2 * k + 0];
    const float px = (float)w + (float)(k % 3 - 1) + offv[2 * k + 1];
    const float y0f = floorf(py), x0f = floorf(px);
    const float wy = py - y0f, wx = px - x0f;
    const int y0 = (int)y0f, x0 = (int)x0f;
    const float w00 = (1.0f - wy) * (1.0f - wx), w01 = (1.0f - wy) * wx;
    const float w10 = wy * (1.0f - wx),          w11 = wy * wx;
    #pragma unroll
    for (int c = 0; c < 2; ++c) {
      const float* cin = c ? ca : cm;
      const float v00 = samp2d(cin, y0,     x0);
      const float v01 = samp2d(cin, y0,     x0 + 1);
      const float v10 = samp2d(cin, y0 + 1, x0);
      const float v11 = samp2d(cin, y0 + 1, x0 + 1);
      acc += (v00 * w00 + v01 * w01 + v10 * w10 + v11 * w11) * s_dw[c * 9 + k];
    }
  }

  const float inv = bn_gamma[0] * rsqrtf(bn_var[0] + 1e-5f);
  const float d   = (acc - bn_mean[0]) * inv + bn_beta[0];
  wsmap[p] = sigmoidf_(d);
}

// ---------------------------------------------------------------------------
// Kernel 5: out = x * (wch*ws + 1)   [ == y*ws + x ]   float4 streaming
// ---------------------------------------------------------------------------
__global__ void __launch_bounds__(256)
fuse_kernel(const float* __restrict__ x, const float* __restrict__ wch,
            const float* __restrict__ wsmap, float* __restrict__ out) {
  const size_t e4 = (size_t)blockIdx.x * 256 + threadIdx.x;   // float4 index
  const size_t e  = e4 * 4;
  const int bc = (int)(e >> 14);          // uniform within a block
  const int b  = bc >> 8;
  const int hw = (int)(e & (HW_ - 1));
  const float scale = wch[bc];
  const float4 xv = reinterpret_cast<const float4*>(x)[e4];
  const float4 wv =
      reinterpret_cast<const float4*>(wsmap + (size_t)b * HW_)[hw >> 2];
  float4 o;
  o.x = xv.x * fmaf(scale, wv.x, 1.0f);
  o.y = xv.y * fmaf(scale, wv.y, 1.0f);
  o.z = xv.z * fmaf(scale, wv.z, 1.0f);
  o.w = xv.w * fmaf(scale, wv.w, 1.0f);
  reinterpret_cast<float4*>(out)[e4] = o;
}

// ---------------------------------------------------------------------------
extern "C" void kernel_launch(void* const* d_in, const int* in_sizes, int n_in,
                              void* d_out, int out_size, void* d_ws, size_t ws_size,
                              hipStream_t stream) {
  (void)in_sizes; (void)n_in; (void)out_size; (void)ws_size;
  const float* x    = (const float*)d_in[0];
  const float* w1   = (const float*)d_in[1];
  const float* b1   = (const float*)d_in[2];
  const float* w2   = (const float*)d_in[3];
  const float* b2   = (const float*)d_in[4];
  const float* offw = (const float*)d_in[5];
  const float* offb = (const float*)d_in[6];
  const float* dcw  = (const float*)d_in[7];
  const float* dcb  = (const float*)d_in[8];
  const float* bng  = (const float*)d_in[9];
  const float* bnb  = (const float*)d_in[10];
  const float* bnm  = (const float*)d_in[11];
  const float* bnv  = (const float*)d_in[12];
  float* out = (float*)d_out;

  // workspace layout (~3.05 MiB total)
  float* pavg  = (float*)d_ws;                    // [16,256]
  float* pmax  = pavg + B_ * C_;                  // [16,256]
  float* wch   = pmax + B_ * C_;                  // [16,256]
  float* cat   = wch + B_ * C_;                   // [16,2,128,128]
  float* wsmap = cat + (size_t)B_ * 2 * HW_;      // [16,128,128]

  pool_kernel<<<B_ * C_, 256, 0, stream>>>(x, pavg, pmax);
  mlp_wmma_kernel<<<1, 32, 0, stream>>>(pavg, pmax, w1, b1, w2, b2, wch);
  ymap_kernel<<<B_ * (HW_ / 256), 256, 0, stream>>>(x, wch, cat);
  dsa_kernel<<<(B_ * HW_) / 256, 256, 0, stream>>>(cat, offw, offb, dcw, dcb,
                                                   bng, bnb, bnm, bnv, wsmap);
  fuse_kernel<<<(int)(((size_t)B_ * C_ * HW_) / 1024), 256, 0, stream>>>(
      x, wch, wsmap, out);
}